// QuantumLayer_15135464751368
// MI455X (gfx1250) — compile-verified
//
#include <hip/hip_runtime.h>
#include <math.h>

// Gaussian Fock-matrix layer, fused:  out = exp(i*kappa*m^2) * (G @ state)
// Single workgroup (latency-bound serial column scan), all state in LDS,
// complex matvec accumulated with V_WMMA_F32_16X16X4_F32 (f32-exact).

typedef __attribute__((ext_vector_type(2))) float v2f;
typedef __attribute__((ext_vector_type(8))) float v8f;

#define CUT 4096
#define NT  1024          // threads per block (32 waves, 1 WGP)
#define EPT (CUT / NT)    // elements per thread per column = 4

struct c32 { float x, y; };
__device__ __forceinline__ c32 cmul(c32 a, c32 b) { return { a.x*b.x - a.y*b.y, a.x*b.y + a.y*b.x }; }
__device__ __forceinline__ c32 cadd(c32 a, c32 b) { return { a.x + b.x, a.y + b.y }; }
__device__ __forceinline__ c32 csub(c32 a, c32 b) { return { a.x - b.x, a.y - b.y }; }
__device__ __forceinline__ c32 cscale(c32 a, float s) { return { a.x*s, a.y*s }; }

__global__ __launch_bounds__(NT) void quantum_gaussian_kernel(
    const float* __restrict__ state_re, const float* __restrict__ state_im,
    const float* __restrict__ p_gr, const float* __restrict__ p_gi,
    const float* __restrict__ p_phi, const float* __restrict__ p_zr,
    const float* __restrict__ p_zi, const float* __restrict__ p_kappa,
    float* __restrict__ out /* interleaved complex64: 2*CUT floats */)
{
    extern __shared__ float smem[];
    // LDS partition (192 KB total, well under 320 KB/WGP):
    float* pan_re = smem;               // [4][CUT] rolling column panel (real)
    float* pan_im = smem + 4 * CUT;     // [4][CUT] rolling column panel (imag)
    float* s_re   = smem + 8 * CUT;     // [CUT] state real
    float* s_im   = smem + 9 * CUT;     // [CUT] state imag
    float* o_re   = smem + 10 * CUT;    // [CUT] accumulated output real
    float* o_im   = smem + 11 * CUT;    // [CUT] accumulated output imag

    const int tid  = threadIdx.x;
    const int lane = tid & 31;
    const int wave = tid >> 5;

    // ---- stage state into LDS (stride-NT => conflict-free) ----
    for (int j = 0; j < EPT; ++j) {
        int m = tid + j * NT;
        s_re[m] = state_re[m];
        s_im[m] = state_im[m];
    }

    // ---- scalar Gaussian parameters (uniform; every lane computes) ----
    const float gr = p_gr[0], gi = p_gi[0], phi = p_phi[0];
    const float zr = p_zr[0], zi = p_zi[0], kappa = p_kappa[0];
    const float r     = sqrtf(zr * zr + zi * zi);
    const float delta = atan2f(zi, zr);
    const float th    = tanhf(r);
    const float sech  = 1.0f / coshf(r);
    const c32 e2 = { cosf(2.0f * phi + delta), sinf(2.0f * phi + delta) };
    const c32 e1 = { cosf(phi), sinf(phi) };
    const c32 gamma = { gr, gi };
    const c32 gconj = { gr, -gi };
    const c32 S00 = cscale(e2, th);                       //  e2 * tanh
    const c32 S01 = cscale(e1, -sech);                    // -e1 * sech
    const c32 S11 = { -th * cosf(delta), th * sinf(delta) }; // -exp(-i d)*tanh
    const c32 mu0 = cadd(gamma, cscale(cmul(gconj, e2), th));
    const c32 mu1 = cscale(cmul(gconj, e1), -sech);
    // C = exp(-0.5|g|^2 - 0.5 conj(g)^2 e2 t) * sqrt(sech)
    c32 gc2 = cmul(gconj, gconj);
    c32 z2  = cscale(cmul(gc2, e2), 0.5f * th);
    float exr = -0.5f * (gr * gr + gi * gi) - z2.x;
    float exi = -z2.y;
    float mag = expf(exr) * sqrtf(sech);
    const c32 C0 = { mag * cosf(exi), mag * sinf(exi) };

    // ---- column 0: strictly serial 2-term recurrence (one thread) ----
    if (tid == 0) {
        c32 cm1 = { 0.f, 0.f };
        c32 cm  = C0;
        pan_re[0] = C0.x; pan_im[0] = C0.y;     // slot 0, m = 0
        for (int m = 1; m < CUT; ++m) {
            float inv = 1.0f / sqrtf((float)m);
            c32 t1 = csub(cmul(mu0, cm), cscale(cmul(S00, cm1), sqrtf((float)(m - 1))));
            c32 cn = cscale(t1, inv);
            pan_re[m] = cn.x; pan_im[m] = cn.y;
            cm1 = cm; cm = cn;
        }
    }
    __syncthreads();

    // ---- persistent WMMA accumulators: 8 m-tiles per wave (rows wave*128 .. +127)
    v8f acc[8];
    for (int i = 0; i < 8; ++i) acc[i] = v8f{0.f,0.f,0.f,0.f,0.f,0.f,0.f,0.f};

    // ---- main loop: groups of 4 columns (panel slot = n & 3) ----
    for (int g = 0; g < CUT / 4; ++g) {
        // (a) generate columns 4g .. 4g+3 in parallel over m
        for (int c = 0; c < 4; ++c) {
            int n = 4 * g + c;
            if (n > 0) {
                int scur = (n - 1) & 3, sprev = (n - 2) & 3, snew = n & 3;
                float snm1 = sqrtf((float)(n - 1));
                float inv  = 1.0f / sqrtf((float)n);
                c32 s11s = cscale(S11, snm1);
                for (int j = 0; j < EPT; ++j) {
                    int m = tid + j * NT;
                    c32 cur   = { pan_re[scur * CUT + m], pan_im[scur * CUT + m] };
                    c32 curm1 = { 0.f, 0.f };
                    if (m > 0) curm1 = { pan_re[scur * CUT + m - 1], pan_im[scur * CUT + m - 1] };
                    c32 prev  = { 0.f, 0.f };
                    if (n >= 2) prev = { pan_re[sprev * CUT + m], pan_im[sprev * CUT + m] };
                    c32 t1 = cmul(mu1, cur);
                    t1 = csub(t1, cscale(cmul(S01, curm1), sqrtf((float)m)));
                    t1 = csub(t1, cmul(s11s, prev));
                    c32 nw = cscale(t1, inv);
                    pan_re[snew * CUT + m] = nw.x;
                    pan_im[snew * CUT + m] = nw.y;
                }
            }
            __syncthreads();
        }

        // (b) accumulate matvec contribution of the 4 columns: two K=4 WMMAs/tile.
        //     A 16x4 (f32) layout: lanes 0-15 -> K=0,1 ; lanes 16-31 -> K=2,3.
        //     K pair (0,1) = complex column na; K pair (2,3) = column na+1:
        //     A = [G_re(na), G_im(na), G_re(na+1), G_im(na+1)]
        //     B col0 = [s_re, -s_im, s_re, -s_im] ; B col1 = [s_im, s_re, s_im, s_re]
        //     => D[:,0] += Re(G*s) ; D[:,1] += Im(G*s)
        for (int ch = 0; ch < 2; ++ch) {
            int na  = 4 * g + 2 * ch;
            int n   = na + ((lane < 16) ? 0 : 1);   // complex column this half-wave feeds
            int slot = n & 3;
            int cidx = lane & 15;                    // B/D column index for this lane
            v2f bfrag = { 0.f, 0.f };
            if (cidx == 0)      { bfrag.x = s_re[n]; bfrag.y = -s_im[n]; }
            else if (cidx == 1) { bfrag.x = s_im[n]; bfrag.y =  s_re[n]; }
            for (int i = 0; i < 8; ++i) {
                int row = (wave * 8 + i) * 16 + cidx;   // A row = M index
                v2f afrag;
                afrag.x = pan_re[slot * CUT + row];
                afrag.y = pan_im[slot * CUT + row];
                acc[i] = __builtin_amdgcn_wmma_f32_16x16x4_f32(
                    /*neg_a=*/false, afrag, /*neg_b=*/false, bfrag,
                    /*c_mod=*/(short)0, acc[i], /*reuse_a=*/false, /*reuse_b=*/false);
            }
        }
        __syncthreads();   // WMMA panel reads done before next group overwrites
    }

    // ---- spill D columns 0 (real) / 1 (imag) to LDS.
    //      D 16x16 layout: VGPR r holds M=r (lanes 0-15) and M=r+8 (lanes 16-31);
    //      lane column = lane & 15.
    for (int i = 0; i < 8; ++i) {
        int m0 = (wave * 8 + i) * 16;
        if (lane == 0)       { for (int rr = 0; rr < 8; ++rr) o_re[m0 + rr]     = acc[i][rr]; }
        else if (lane == 16) { for (int rr = 0; rr < 8; ++rr) o_re[m0 + 8 + rr] = acc[i][rr]; }
        else if (lane == 1)  { for (int rr = 0; rr < 8; ++rr) o_im[m0 + rr]     = acc[i][rr]; }
        else if (lane == 17) { for (int rr = 0; rr < 8; ++rr) o_im[m0 + 8 + rr] = acc[i][rr]; }
    }
    __syncthreads();

    // ---- Kerr phase + store interleaved complex64 ----
    for (int j = 0; j < EPT; ++j) {
        int m = tid + j * NT;
        float fm  = (float)m;
        float ang = kappa * fm * fm;     // m^2 < 2^24 -> exact in f32 (matches ref)
        float sn, cs;
        __sincosf(ang, &sn, &cs);
        float re = o_re[m], im = o_im[m];
        out[2 * m]     = cs * re - sn * im;
        out[2 * m + 1] = cs * im + sn * re;
    }
}

extern "C" void kernel_launch(void* const* d_in, const int* in_sizes, int n_in,
                              void* d_out, int out_size, void* d_ws, size_t ws_size,
                              hipStream_t stream) {
    (void)in_sizes; (void)n_in; (void)d_ws; (void)ws_size; (void)out_size;
    const size_t shmem = (size_t)12 * CUT * sizeof(float);   // 192 KB LDS
    quantum_gaussian_kernel<<<1, NT, shmem, stream>>>(
        (const float*)d_in[0], (const float*)d_in[1],
        (const float*)d_in[2], (const float*)d_in[3], (const float*)d_in[4],
        (const float*)d_in[5], (const float*)d_in[6], (const float*)d_in[7],
        (float*)d_out);
}